// StateConditionedRetriever_41455024341253
// MI455X (gfx1250) — compile-verified
//
#include <hip/hip_runtime.h>
#include <hip/hip_bf16.h>
#include <math.h>

// ---------------- problem constants (from reference setup_inputs) -----------
#define BATCH     512       // rows of state / query
#define D_STATE   512
#define D_CHUNK   384
#define N_CHUNKS  262144
#define TOPK      32
#define NBLK      512                   // chunk columns per workgroup
#define NPART     (N_CHUNKS / NBLK)     // 512 partitions
#define NTILES    (NBLK / 16)           // 32 N-tiles of 16 per workgroup
#define CH_STRIDE 388                   // padded LDS row stride (floats): 388%64==4 -> conflict-free
#define TILE_F    (16 * CH_STRIDE)      // floats per LDS chunk tile buffer
#define SC_STRIDE 17                    // padded score-tile stride
#define NEG_INF   (-1.0e30f)

typedef float v2f __attribute__((ext_vector_type(2)));
typedef float v8f __attribute__((ext_vector_type(8)));
typedef int   vsi4 __attribute__((vector_size(16)));   // matches builtin param type

// CDNA5 async memory->LDS path (ASYNCcnt, decoupled from LOADcnt)
#if defined(__gfx1250__) && \
    __has_builtin(__builtin_amdgcn_global_load_async_to_lds_b128) && \
    __has_builtin(__builtin_amdgcn_s_wait_asynccnt)
#define USE_ASYNC_LDS 1
#define ASYNC_B128(gp, lp, off) \
  __builtin_amdgcn_global_load_async_to_lds_b128((vsi4*)(gp), (vsi4*)(lp), (off), 0)
#endif

// ---------------- phase 1: query head ---------------------------------------
__global__ void __launch_bounds__(128) retr_mlp1(const float* __restrict__ state,
                                                 const float* __restrict__ w1,
                                                 const float* __restrict__ b1,
                                                 float* __restrict__ h) {
  const int j = blockIdx.y * 128 + threadIdx.x;   // 0..383
  const int b = blockIdx.x;                       // 0..511
  float acc = b1[j];
  const float* s = state + b * D_STATE;
  #pragma unroll 4
  for (int k = 0; k < D_STATE; ++k) acc = fmaf(s[k], w1[k * D_CHUNK + j], acc);
  h[b * D_CHUNK + j] = fmaxf(acc, 0.0f);
}

__global__ void __launch_bounds__(128) retr_mlp2(const float* __restrict__ h,
                                                 const float* __restrict__ w2,
                                                 const float* __restrict__ b2,
                                                 float* __restrict__ q) {
  const int j = blockIdx.y * 128 + threadIdx.x;
  const int b = blockIdx.x;
  float acc = b2[j];
  const float* hr = h + b * D_CHUNK;
  #pragma unroll 4
  for (int k = 0; k < D_CHUNK; ++k) acc = fmaf(hr[k], w2[k * D_CHUNK + j], acc);
  q[b * D_CHUNK + j] = acc;
}

__global__ void __launch_bounds__(128) retr_l2norm(float* __restrict__ q) {
  __shared__ float red[128];
  const int b = blockIdx.x, t = threadIdx.x;
  float ss = 0.0f;
  for (int j = t; j < D_CHUNK; j += 128) { float v = q[b * D_CHUNK + j]; ss += v * v; }
  red[t] = ss; __syncthreads();
  for (int s = 64; s > 0; s >>= 1) { if (t < s) red[t] += red[t + s]; __syncthreads(); }
  const float inv = 1.0f / fmaxf(sqrtf(red[0]), 1e-12f);
  for (int j = t; j < D_CHUNK; j += 128) q[b * D_CHUNK + j] *= inv;
}

// ---------------- phase 2: fused GEMM + chunk-norm + partial top-k ----------
// grid: NPART workgroups of 256 threads (8 waves). Workgroup p scores chunk
// columns [p*512, p*512+512) against all 512 query rows via WMMA f32 16x16x4.
// Chunk tiles are double-buffered in LDS and (when available) streamed with
// GLOBAL_LOAD_ASYNC_TO_LDS_B128 one tile ahead, synchronized via ASYNCcnt.
// NOTE: buffer selection is pure pointer arithmetic on the smem base so that
// infer-address-spaces keeps all tile reads on the DS path (no flat_load).
__global__ void __launch_bounds__(256) retr_gemm_topk(const float* __restrict__ q,
                                                      const float* __restrict__ chunks,
                                                      float* __restrict__ candS,
                                                      int*   __restrict__ candI) {
  extern __shared__ float smem[];
  float* ldsChunk  = smem;                              // 2 * 16*388 = 12416 f
  float* ldsScore  = ldsChunk + 2 * TILE_F;             // 512*17 = 8704 f
  float* topS      = ldsScore + BATCH * SC_STRIDE;      // 512*32 = 16384 f
  int*   topI      = (int*)(topS + BATCH * TOPK);       // 512*32 = 16384 i
  float* nacc      = (float*)(topI + BATCH * TOPK);     // 16 f
  float* invn      = nacc + 16;                         // 16 f

  const int tid   = threadIdx.x;
  const int wave  = tid >> 5;
  const int lane  = tid & 31;
  const int lcol  = lane & 15;        // N / M position within tile
  const int lhalf = lane >> 4;        // 0 or 1
  const int blk   = blockIdx.x;

  // staging role: 16 threads per chunk row
  const int sc_c = tid >> 4;          // chunk row 0..15
  const int sc_u = tid & 15;          // float4 slot within row

#ifdef USE_ASYNC_LDS
  { // prologue: stream tile 0 into buffer 0 (6 x B128 per thread-slot)
    const float* g = chunks + (size_t)(blk * NBLK + sc_c) * D_CHUNK + sc_u * 4;
    float*       l = ldsChunk + sc_c * CH_STRIDE + sc_u * 4;
    ASYNC_B128(g, l,    0);
    ASYNC_B128(g, l,  256);
    ASYNC_B128(g, l,  512);
    ASYNC_B128(g, l,  768);
    ASYNC_B128(g, l, 1024);
    ASYNC_B128(g, l, 1280);
  }
#endif

  // init per-row top lists (thread t exclusively owns rows 2t and 2t+1)
  for (int i = tid; i < BATCH * TOPK; i += 256) { topS[i] = NEG_INF; topI[i] = 0; }
  float rmin[2] = {NEG_INF, NEG_INF};
  int   rpos[2] = {0, 0};

  // A-fragment base pointers: wave handles M-tiles wave*4 .. wave*4+3
  const float* qb[4];
  #pragma unroll
  for (int mt = 0; mt < 4; ++mt)
    qb[mt] = q + (size_t)(wave * 64 + mt * 16 + lcol) * D_CHUNK + 2 * lhalf;

  for (int nt = 0; nt < NTILES; ++nt) {
    // buffer select via arithmetic on the smem-rooted pointer (keeps AS(3))
    float* buf = ldsChunk + (nt & 1) * TILE_F;

#ifdef USE_ASYNC_LDS
    // stream tile nt+1 into the other buffer, then wait for tile nt
    if (nt + 1 < NTILES) {
      const float* g = chunks + (size_t)(blk * NBLK + (nt + 1) * 16 + sc_c) * D_CHUNK + sc_u * 4;
      float*       l = ldsChunk + ((nt + 1) & 1) * TILE_F + sc_c * CH_STRIDE + sc_u * 4;
      ASYNC_B128(g, l,    0);
      ASYNC_B128(g, l,  256);
      ASYNC_B128(g, l,  512);
      ASYNC_B128(g, l,  768);
      ASYNC_B128(g, l, 1024);
      ASYNC_B128(g, l, 1280);
      __builtin_amdgcn_s_wait_asynccnt(6);
    } else {
      __builtin_amdgcn_s_wait_asynccnt(0);
    }
#else
    { // synchronous staging fallback
      const float4* src = (const float4*)(chunks +
          (size_t)(blk * NBLK + nt * 16 + sc_c) * D_CHUNK);
      float* dst = buf + sc_c * CH_STRIDE;
      #pragma unroll
      for (int j = 0; j < 6; ++j) {
        const int kq = sc_u + j * 16;
        *(float4*)(dst + kq * 4) = src[kq];
      }
    }
#endif
    if (tid < 16) nacc[tid] = 0.0f;
    __syncthreads();      // tile nt visible to all waves; nacc zeroed

    // ---- per-chunk inverse norms from the staged tile
    {
      const float4* r = (const float4*)(buf + sc_c * CH_STRIDE);
      float ss = 0.0f;
      #pragma unroll
      for (int j = 0; j < 6; ++j) {
        float4 v = r[sc_u + j * 16];
        ss += v.x * v.x + v.y * v.y + v.z * v.z + v.w * v.w;
      }
      atomicAdd(&nacc[sc_c], ss);             // ds_add_f32
    }
    __syncthreads();
    if (tid < 16) invn[tid] = 1.0f / fmaxf(sqrtf(nacc[tid]), 1e-12f);
    __syncthreads();

    // ---- WMMA: 4 M-tiles per wave, K = 384 in steps of 4
    v8f acc0 = {}, acc1 = {}, acc2 = {}, acc3 = {};
    const float* bsrc = buf + lcol * CH_STRIDE + 2 * lhalf;
    for (int k = 0; k < D_CHUNK; k += 4) {
      v2f bfrag = *(const v2f*)(bsrc + k);                    // ds_load_b64
      v2f a0 = *(const v2f*)(qb[0] + k);
      v2f a1 = *(const v2f*)(qb[1] + k);
      v2f a2 = *(const v2f*)(qb[2] + k);
      v2f a3 = *(const v2f*)(qb[3] + k);
      acc0 = __builtin_amdgcn_wmma_f32_16x16x4_f32(false, a0, false, bfrag, (short)0, acc0, false, false);
      acc1 = __builtin_amdgcn_wmma_f32_16x16x4_f32(false, a1, false, bfrag, (short)0, acc1, false, false);
      acc2 = __builtin_amdgcn_wmma_f32_16x16x4_f32(false, a2, false, bfrag, (short)0, acc2, false, false);
      acc3 = __builtin_amdgcn_wmma_f32_16x16x4_f32(false, a3, false, bfrag, (short)0, acc3, false, false);
    }

    // ---- epilogue: scale by chunk inverse norm, stage scores in LDS
    const float sc = invn[lcol];
    const int rb = wave * 64 + lhalf * 8;     // C layout: M = vgpr + 8*lhalf
    #pragma unroll
    for (int i = 0; i < 8; ++i) {
      ldsScore[(rb +  0 + i) * SC_STRIDE + lcol] = acc0[i] * sc;
      ldsScore[(rb + 16 + i) * SC_STRIDE + lcol] = acc1[i] * sc;
      ldsScore[(rb + 32 + i) * SC_STRIDE + lcol] = acc2[i] * sc;
      ldsScore[(rb + 48 + i) * SC_STRIDE + lcol] = acc3[i] * sc;
    }
    __syncthreads();

    // ---- per-row top-32 maintenance (thread t owns rows 2t, 2t+1)
    const int gbase = blk * NBLK + nt * 16;
    #pragma unroll
    for (int rr = 0; rr < 2; ++rr) {
      const int r = tid * 2 + rr;
      float mn = rmin[rr];
      int   mp = rpos[rr];
      float* ts = topS + r * TOPK;
      int*   ti = topI + r * TOPK;
      for (int c = 0; c < 16; ++c) {
        const float s = ldsScore[r * SC_STRIDE + c];
        if (s > mn) {                          // strict: earlier (lower) index wins ties
          ts[mp] = s; ti[mp] = gbase + c;
          mn = ts[0]; mp = 0;
          for (int j = 1; j < TOPK; ++j)
            if (ts[j] < mn) { mn = ts[j]; mp = j; }
        }
      }
      rmin[rr] = mn; rpos[rr] = mp;
    }
    __syncthreads();      // lists stable; tile buffer (nt&1) free for reuse
  }

  // ---- emit this partition's candidates
  for (int i = tid; i < BATCH * TOPK; i += 256) {
    candS[(size_t)blk * BATCH * TOPK + i] = topS[i];
    candI[(size_t)blk * BATCH * TOPK + i] = topI[i];
  }
}

// ---------------- phase 3: merge partitions -> global top-32 per row --------
__global__ void __launch_bounds__(256) retr_merge_topk(const float* __restrict__ candS,
                                                       const int*   __restrict__ candI,
                                                       int*   __restrict__ outI,
                                                       float* __restrict__ outS) {
  extern __shared__ float smem[];
  const int NC = NPART * TOPK;                // 16384 candidates per row
  float* sc   = smem;                         // 16384 f
  int*   id   = (int*)(sc + NC);              // 16384 i
  float* redV = (float*)(id + NC);            // 256 f
  int*   redI = (int*)(redV + 256);           // 256 i
  int*   redP = (int*)(redI + 256);           // 256 i

  const int b = blockIdx.x, t = threadIdx.x;
  for (int i = t; i < NC; i += 256) {
    const size_t src = ((size_t)(i >> 5) * BATCH + b) * TOPK + (i & 31);
    sc[i] = candS[src];
    id[i] = candI[src];
  }
  __syncthreads();

  for (int r = 0; r < TOPK; ++r) {
    float bv = NEG_INF; int bi = 0x7fffffff, bp = 0;
    for (int i = t; i < NC; i += 256) {
      const float v = sc[i];
      if (v > bv || (v == bv && id[i] < bi)) { bv = v; bi = id[i]; bp = i; }
    }
    redV[t] = bv; redI[t] = bi; redP[t] = bp;
    __syncthreads();
    for (int s = 128; s > 0; s >>= 1) {
      if (t < s) {
        const float ov = redV[t + s]; const int oi = redI[t + s];
        if (ov > redV[t] || (ov == redV[t] && oi < redI[t])) {
          redV[t] = ov; redI[t] = oi; redP[t] = redP[t + s];
        }
      }
      __syncthreads();
    }
    if (t == 0) {
      outI[b * TOPK + r] = redI[0];
      outS[b * TOPK + r] = redV[0];
      sc[redP[0]] = NEG_INF;                  // remove winner
    }
    __syncthreads();
  }
}

// ---------------- launch -----------------------------------------------------
extern "C" void kernel_launch(void* const* d_in, const int* in_sizes, int n_in,
                              void* d_out, int out_size, void* d_ws, size_t ws_size,
                              hipStream_t stream) {
  const float* state  = (const float*)d_in[0];
  const float* chunks = (const float*)d_in[1];
  const float* w1     = (const float*)d_in[2];
  const float* b1     = (const float*)d_in[3];
  const float* w2     = (const float*)d_in[4];
  const float* b2     = (const float*)d_in[5];
  // d_in[6] is top_k == 32 (hardcoded as TOPK)

  char* ws = (char*)d_ws;
  const size_t candElems = (size_t)NPART * BATCH * TOPK;       // 8,388,608
  float* candS = (float*)ws;
  int*   candI = (int*)(ws + candElems * sizeof(float));
  float* h     = (float*)(ws + 2 * candElems * sizeof(float));
  float* q     = h + BATCH * D_CHUNK;

  retr_mlp1<<<dim3(BATCH, D_CHUNK / 128), 128, 0, stream>>>(state, w1, b1, h);
  retr_mlp2<<<dim3(BATCH, D_CHUNK / 128), 128, 0, stream>>>(h, w2, b2, q);
  retr_l2norm<<<BATCH, 128, 0, stream>>>(q);

  const size_t lds4 = (size_t)(2 * TILE_F + BATCH * SC_STRIDE + BATCH * TOPK) * 4
                    + (size_t)BATCH * TOPK * 4 + 32 * 4;       // ~211 KB
  retr_gemm_topk<<<NPART, 256, lds4, stream>>>(q, chunks, candS, candI);

  const size_t lds5 = (size_t)NPART * TOPK * 8 + 256 * 12;     // ~131 KB
  retr_merge_topk<<<BATCH, 256, lds5, stream>>>(candS, candI,
                                                (int*)d_out,
                                                (float*)d_out + BATCH * TOPK);
}